// GATConv_62182536511744
// MI455X (gfx1250) — compile-verified
//
#include <hip/hip_runtime.h>
#include <math.h>

#define NNODES 100000
#define NEDGES 1600000
#define DIN    256
#define DOUT   32
#define ALPHA  0.2f
#define NBLK3  1024

typedef __attribute__((ext_vector_type(2))) float v2f;
typedef __attribute__((ext_vector_type(8))) float v8f;

// ---------------------------------------------------------------------------
// Kernel 1: Xp = X @ W via V_WMMA_F32_16X16X4_F32.
// One wave -> one 16-row tile across BOTH 16-col tiles (two accumulators),
// so each A fragment is loaded once and feeds two WMMAs. Block = 8 waves.
//
// 32-bit A 16x4 layout: lanes 0-15 {V0=K0, V1=K1}, lanes 16-31 {V0=K2, V1=K3}
// 32-bit B 4x16 layout: lane%16 = column N; same K striping as A.
// 32-bit C/D 16x16:     VGPR v -> row (v + 8*(lane/16)), col = lane%16.
// ---------------------------------------------------------------------------
__global__ __launch_bounds__(256) void gat_gemm_wmma(
    const float* __restrict__ X, const float* __restrict__ W,
    float* __restrict__ Xp) {
  const int lane = threadIdx.x & 31;
  const int wave = threadIdx.x >> 5;
  const int m0   = (blockIdx.x * 8 + wave) << 4;   // row tile base
  if (m0 >= NNODES) return;                        // wave-uniform: EXEC stays ~0
  const int mr = lane & 15;
  const int h  = lane >> 4;                        // lane-half -> K pair select

  const float* __restrict__ arow  = X + (size_t)(m0 + mr) * DIN;
  const float* __restrict__ bcol0 = W + mr;        // columns 0..15
  const float* __restrict__ bcol1 = W + 16 + mr;   // columns 16..31

  v8f c0 = {};
  v8f c1 = {};
#pragma unroll 4
  for (int k = 0; k < DIN; k += 4) {
    const int kk = k + 2 * h;
    v2f a;
    a.x = arow[kk];
    a.y = arow[kk + 1];                            // contiguous -> b64 load
    v2f b0, b1;
    b0.x = bcol0[(size_t)kk * DOUT];
    b0.y = bcol0[(size_t)(kk + 1) * DOUT];
    b1.x = bcol1[(size_t)kk * DOUT];
    b1.y = bcol1[(size_t)(kk + 1) * DOUT];
    c0 = __builtin_amdgcn_wmma_f32_16x16x4_f32(false, a, false, b0, (short)0,
                                               c0, false, false);
    c1 = __builtin_amdgcn_wmma_f32_16x16x4_f32(false, a, false, b1, (short)0,
                                               c1, false, false);
  }

  float* __restrict__ o = Xp + (size_t)(m0 + 8 * h) * DOUT + mr;
#pragma unroll
  for (int v = 0; v < 8; ++v) {
    o[(size_t)v * DOUT]      = c0[v];
    o[(size_t)v * DOUT + 16] = c1[v];
  }
}

// ---------------------------------------------------------------------------
// Kernel 2: s0[n] = Xp[n,:].a0 ; s1[n] = Xp[n,:].a1   (one thread per node)
// ---------------------------------------------------------------------------
__global__ __launch_bounds__(256) void gat_scores(
    const float* __restrict__ Xp, const float* __restrict__ a0,
    const float* __restrict__ a1, float* __restrict__ s0,
    float* __restrict__ s1) {
  const int n = blockIdx.x * 256 + threadIdx.x;
  if (n >= NNODES) return;
  const float4* __restrict__ r  = (const float4*)(Xp + (size_t)n * DOUT);
  const float4* __restrict__ A0 = (const float4*)a0;
  const float4* __restrict__ A1 = (const float4*)a1;
  float acc0 = 0.f, acc1 = 0.f;
#pragma unroll
  for (int i = 0; i < DOUT / 4; ++i) {
    const float4 x = r[i], p = A0[i], q = A1[i];
    acc0 += x.x * p.x + x.y * p.y + x.z * p.z + x.w * p.w;
    acc1 += x.x * q.x + x.y * q.y + x.z * q.z + x.w * q.w;
  }
  s0[n] = acc0;
  s1[n] = acc1;
}

// ---------------------------------------------------------------------------
// Kernel 3: raw edge scores (leaky-relu) + per-block min/max partials.
// ---------------------------------------------------------------------------
__global__ __launch_bounds__(256) void gat_edge_minmax(
    const float* __restrict__ s0, const float* __restrict__ s1,
    const int* __restrict__ row, const int* __restrict__ col,
    float* __restrict__ att_raw, float* __restrict__ pmin,
    float* __restrict__ pmax) {
  __shared__ float smin[256];
  __shared__ float smax[256];
  float lmin = INFINITY, lmax = -INFINITY;
  for (int e = blockIdx.x * 256 + threadIdx.x; e < NEDGES;
       e += gridDim.x * 256) {
    float v = s0[row[e]] + s1[col[e]];
    v = v > 0.f ? v : ALPHA * v;
    att_raw[e] = v;
    lmin = fminf(lmin, v);
    lmax = fmaxf(lmax, v);
  }
  smin[threadIdx.x] = lmin;
  smax[threadIdx.x] = lmax;
  __syncthreads();
  for (int s = 128; s > 0; s >>= 1) {
    if (threadIdx.x < (unsigned)s) {
      smin[threadIdx.x] = fminf(smin[threadIdx.x], smin[threadIdx.x + s]);
      smax[threadIdx.x] = fmaxf(smax[threadIdx.x], smax[threadIdx.x + s]);
    }
    __syncthreads();
  }
  if (threadIdx.x == 0) {
    pmin[blockIdx.x] = smin[0];
    pmax[blockIdx.x] = smax[0];
  }
}

// ---------------------------------------------------------------------------
// Kernel 4: final reduction -> scal[0]=mn, scal[1]=1/(mx-mn)
// ---------------------------------------------------------------------------
__global__ __launch_bounds__(256) void gat_minmax_final(
    const float* __restrict__ pmin, const float* __restrict__ pmax,
    float* __restrict__ scal) {
  __shared__ float smin[256];
  __shared__ float smax[256];
  float lmin = INFINITY, lmax = -INFINITY;
  for (int i = threadIdx.x; i < NBLK3; i += 256) {
    lmin = fminf(lmin, pmin[i]);
    lmax = fmaxf(lmax, pmax[i]);
  }
  smin[threadIdx.x] = lmin;
  smax[threadIdx.x] = lmax;
  __syncthreads();
  for (int s = 128; s > 0; s >>= 1) {
    if (threadIdx.x < (unsigned)s) {
      smin[threadIdx.x] = fminf(smin[threadIdx.x], smin[threadIdx.x + s]);
      smax[threadIdx.x] = fmaxf(smax[threadIdx.x], smax[threadIdx.x + s]);
    }
    __syncthreads();
  }
  if (threadIdx.x == 0) {
    scal[0] = smin[0];
    scal[1] = 1.f / (smax[0] - smin[0]);
  }
}

// ---------------------------------------------------------------------------
// Kernel 5: row_start[n] = lower_bound(row_idx, n); row_idx is sorted.
// Computes CSR-style segment offsets; start[NNODES] == NEDGES.
// ---------------------------------------------------------------------------
__global__ __launch_bounds__(256) void gat_row_start(
    const int* __restrict__ row, int* __restrict__ start) {
  const int n = blockIdx.x * 256 + threadIdx.x;
  if (n > NNODES) return;
  int lo = 0, hi = NEDGES;
  while (lo < hi) {
    const int mid = (lo + hi) >> 1;
    if (row[mid] < n)
      lo = mid + 1;
    else
      hi = mid;
  }
  start[n] = lo;
}

// ---------------------------------------------------------------------------
// Kernel 6: segmented (atomic-free) aggregation + softmax-normalize.
// One wave per destination node; lane = output column. Edges of node n are
// the contiguous range [start[n], start[n+1]) because row_idx is sorted.
// out[n,:] = (sum_e att_e * Xp[col_e,:]) / (sum_e att_e)
// Empty segment -> 0/0 = NaN, matching the reference's segment_sum semantics.
// ---------------------------------------------------------------------------
__global__ __launch_bounds__(256) void gat_aggregate(
    const float* __restrict__ att_raw, const int* __restrict__ col,
    const int* __restrict__ start, const float* __restrict__ Xp,
    const float* __restrict__ scal, float* __restrict__ out) {
  const int n    = (blockIdx.x * 256 + threadIdx.x) >> 5;  // node id
  const int lane = threadIdx.x & 31;
  if (n >= NNODES) return;
  const float mn  = scal[0];
  const float inv = scal[1];
  const int e0 = start[n];
  const int e1 = start[n + 1];
  float acc = 0.f, asum = 0.f;
  for (int e = e0; e < e1; ++e) {
    const float att = expf((att_raw[e] - mn) * inv);
    acc  = fmaf(att, Xp[(size_t)col[e] * DOUT + lane], acc);
    asum += att;
  }
  out[(size_t)n * DOUT + lane] = acc / asum;
}

// ---------------------------------------------------------------------------
extern "C" void kernel_launch(void* const* d_in, const int* in_sizes, int n_in,
                              void* d_out, int out_size, void* d_ws,
                              size_t ws_size, hipStream_t stream) {
  const float* X   = (const float*)d_in[0];
  const float* W   = (const float*)d_in[1];
  const float* a0  = (const float*)d_in[2];
  const float* a1  = (const float*)d_in[3];
  const int*   row = (const int*)d_in[4];
  const int*   col = (const int*)d_in[5];
  float* out = (float*)d_out;

  float* ws      = (float*)d_ws;
  float* Xp      = ws;                          // N*DOUT
  float* s0      = Xp + (size_t)NNODES * DOUT;  // N
  float* s1      = s0 + NNODES;                 // N
  float* att_raw = s1 + NNODES;                 // E
  float* pmin    = att_raw + NEDGES;            // NBLK3
  float* pmax    = pmin + NBLK3;                // NBLK3
  float* scal    = pmax + NBLK3;                // 2
  int*   start   = (int*)(scal + 2);            // N+1

  const int gemm_blocks = (NNODES / 16 + 7) / 8;  // 8 row-tiles per block
  gat_gemm_wmma<<<gemm_blocks, 256, 0, stream>>>(X, W, Xp);
  gat_scores<<<(NNODES + 255) / 256, 256, 0, stream>>>(Xp, a0, a1, s0, s1);
  gat_edge_minmax<<<NBLK3, 256, 0, stream>>>(s0, s1, row, col, att_raw, pmin,
                                             pmax);
  gat_minmax_final<<<1, 256, 0, stream>>>(pmin, pmax, scal);
  gat_row_start<<<(NNODES + 1 + 255) / 256, 256, 0, stream>>>(row, start);
  gat_aggregate<<<(NNODES * 32 + 255) / 256, 256, 0, stream>>>(
      att_raw, col, start, Xp, scal, out);
}